// VectorQuantizer_71957882077238
// MI455X (gfx1250) — compile-verified
//
#include <hip/hip_runtime.h>

typedef __attribute__((ext_vector_type(2))) float f32x2;
typedef __attribute__((ext_vector_type(8))) float f32x8;

#define D_DIM 256
#define K_EMB 8192
#define N_ROWS 16384
#define ROWS_PER_WG 128
#define WG_THREADS 256
#define LDS_STRIDE 260              // 256 + 4 pad dwords: conflict-free b64 frags, 16B-aligned rows
#define NTILES (K_EMB / 16)         // 512 column tiles

#if defined(__HIP_DEVICE_COMPILE__) && __has_builtin(__builtin_amdgcn_global_load_async_to_lds_b128)
#define HAVE_ASYNC_LDS 1
typedef int v4i_ __attribute__((vector_size(16)));
typedef __attribute__((address_space(1))) v4i_ GV4;   // global-AS 4xi32
typedef __attribute__((address_space(3))) v4i_ LV4;   // LDS-AS    4xi32
#if __has_builtin(__builtin_amdgcn_s_wait_asynccnt)
#define WAIT_ASYNC() __builtin_amdgcn_s_wait_asynccnt(0)
#else
#define WAIT_ASYNC() asm volatile("s_wait_asynccnt 0x0" ::: "memory")
#endif
#else
#define HAVE_ASYNC_LDS 0
#endif

// Workspace layout (bytes):
//   [0,      32768)  : enorm   float[8192]
//   [32768,  98304)  : indices int[16384]
//   [98304, 131072)  : counts  uint[8192]
//   [131072,131080)  : lossAcc double
//
// d_out: quantized_st (16384*256 f32) | loss (f32) | perplexity (f32)

__global__ void vq_init_ws(unsigned* counts, double* lossAcc) {
    int t = blockIdx.x * blockDim.x + threadIdx.x;
    if (t < K_EMB) counts[t] = 0u;
    if (t == 0) *lossAcc = 0.0;
}

__global__ __launch_bounds__(256) void vq_enorm(const float* __restrict__ E,
                                                float* __restrict__ enorm) {
    int wave = threadIdx.x >> 5, lane = threadIdx.x & 31;
    int row = blockIdx.x * 8 + wave;
    const float4* p = (const float4*)(E + (size_t)row * D_DIM);
    float4 a = p[lane];
    float4 b = p[lane + 32];
    float s = a.x * a.x + a.y * a.y + a.z * a.z + a.w * a.w
            + b.x * b.x + b.y * b.y + b.z * b.z + b.w * b.w;
#pragma unroll
    for (int off = 16; off > 0; off >>= 1) s += __shfl_xor(s, off, 32);
    if (lane == 0) enorm[row] = s;
}

__global__ __launch_bounds__(WG_THREADS, 2) void vq_argmin(
    const float* __restrict__ x, const float* __restrict__ E,
    const float* __restrict__ enorm, int* __restrict__ outIdx) {
    extern __shared__ float smem[];
    float* sA  = smem;                              // 128 x 260
    float* sB0 = sA + ROWS_PER_WG * LDS_STRIDE;     // 16 x 260, double-buffered
    float* sB1 = sB0 + 16 * LDS_STRIDE;

    const int tid  = threadIdx.x;
    const int wave = tid >> 5;
    const int lane = tid & 31;
    const int r    = lane & 15;                     // M (A) / N (B) index
    const int koff = (lane >> 4) << 1;              // lanes 16-31 take K+2,K+3
    const int wgRow0 = blockIdx.x * ROWS_PER_WG;

#if HAVE_ASYNC_LDS
    // Stage x tile (128x256) + embedding tile 0 via async DMA: global -> LDS, no VGPR staging.
    {
        const float* gx = x + (size_t)wgRow0 * D_DIM;
#pragma unroll
        for (int j = 0; j < 32; ++j) {
            int c = tid + j * 256;                  // 8192 b128 chunks
            int row = c >> 6, col4 = c & 63;
            __builtin_amdgcn_global_load_async_to_lds_b128(
                (GV4*)(gx + row * D_DIM + col4 * 4),
                (LV4*)(sA + row * LDS_STRIDE + col4 * 4), 0, 0);
        }
#pragma unroll
        for (int j = 0; j < 4; ++j) {
            int c = tid + j * 256;                  // 1024 b128 chunks
            int row = c >> 6, col4 = c & 63;
            __builtin_amdgcn_global_load_async_to_lds_b128(
                (GV4*)(E + row * D_DIM + col4 * 4),
                (LV4*)(sB0 + row * LDS_STRIDE + col4 * 4), 0, 0);
        }
        WAIT_ASYNC();
    }
#else
    {
        const float4* gx = (const float4*)(x + (size_t)wgRow0 * D_DIM);
#pragma unroll
        for (int j = 0; j < 32; ++j) {
            int c = tid + j * 256;
            int row = c >> 6, col4 = c & 63;
            float4 v = gx[row * 64 + col4];
            *(float4*)(sA + row * LDS_STRIDE + col4 * 4) = v;
        }
        const float4* ge = (const float4*)E;
#pragma unroll
        for (int j = 0; j < 4; ++j) {
            int c = tid + j * 256;
            int row = c >> 6, col4 = c & 63;
            *(float4*)(sB0 + row * LDS_STRIDE + col4 * 4) = ge[row * 64 + col4];
        }
    }
#endif
    __syncthreads();

    float best[8];
    int   bidx[8];
#pragma unroll
    for (int i = 0; i < 8; ++i) { best[i] = 3.4e38f; bidx[i] = 0; }

    const float* pa = sA + (wave * 16 + r) * LDS_STRIDE + koff;

    for (int kt = 0; kt < NTILES; ++kt) {
        float* sBc = (kt & 1) ? sB1 : sB0;
        float* sBn = (kt & 1) ? sB0 : sB1;

#if HAVE_ASYNC_LDS
        // Async-DMA the next embedding tile into the other buffer; overlaps the WMMA chain.
        // Safe: buffer sBn was last read in the previous iteration, fenced by the previous barrier.
        if (kt + 1 < NTILES) {
            const float* ge = E + (size_t)(kt + 1) * 16 * D_DIM;
#pragma unroll
            for (int j = 0; j < 4; ++j) {
                int c = tid + j * 256;
                int row = c >> 6, col4 = c & 63;
                __builtin_amdgcn_global_load_async_to_lds_b128(
                    (GV4*)(ge + row * D_DIM + col4 * 4),
                    (LV4*)(sBn + row * LDS_STRIDE + col4 * 4), 0, 0);
            }
        }
#else
        float4 rv[4];
        if (kt + 1 < NTILES) {
            const float4* ge = (const float4*)(E + (size_t)(kt + 1) * 16 * D_DIM);
#pragma unroll
            for (int j = 0; j < 4; ++j) {
                int c = tid + j * 256;
                rv[j] = ge[(c >> 6) * 64 + (c & 63)];
            }
        }
#endif
        float en = enorm[kt * 16 + r];

        const float* pb = sBc + r * LDS_STRIDE + koff;
        f32x8 acc0 = {0.f, 0.f, 0.f, 0.f, 0.f, 0.f, 0.f, 0.f};
        f32x8 acc1 = {0.f, 0.f, 0.f, 0.f, 0.f, 0.f, 0.f, 0.f};
        // 64 K-steps of 16x16x4 fp32 WMMA, two interleaved chains.
#pragma unroll
        for (int t = 0; t < 64; t += 2) {
            f32x2 a0 = *(const f32x2*)(pa + 4 * t);
            f32x2 b0 = *(const f32x2*)(pb + 4 * t);
            acc0 = __builtin_amdgcn_wmma_f32_16x16x4_f32(
                false, a0, false, b0, (short)0, acc0, false, false);
            f32x2 a1 = *(const f32x2*)(pa + 4 * t + 4);
            f32x2 b1 = *(const f32x2*)(pb + 4 * t + 4);
            acc1 = __builtin_amdgcn_wmma_f32_16x16x4_f32(
                false, a1, false, b1, (short)0, acc1, false, false);
        }

        // dist = ||e||^2 - 2*dot  (||x||^2 constant per row; dropped)
        int n = kt * 16 + r;
#pragma unroll
        for (int i = 0; i < 8; ++i) {
            float d = en - 2.0f * (acc0[i] + acc1[i]);
            if (d < best[i]) { best[i] = d; bidx[i] = n; }   // strict < keeps lowest index
        }

#if HAVE_ASYNC_LDS
        WAIT_ASYNC();
#else
        if (kt + 1 < NTILES) {
#pragma unroll
            for (int j = 0; j < 4; ++j) {
                int c = tid + j * 256;
                *(float4*)(sBn + (c >> 6) * LDS_STRIDE + (c & 63) * 4) = rv[j];
            }
        }
#endif
        __syncthreads();
    }

    // Cross-lane argmin per row. C layout: VGPR i -> M=i (lanes 0-15), M=8+i (lanes 16-31).
#pragma unroll
    for (int i = 0; i < 8; ++i) {
        float d = best[i];
        int   ix = bidx[i];
#pragma unroll
        for (int off = 8; off > 0; off >>= 1) {
            float od = __shfl_xor(d, off, 16);
            int   oi = __shfl_xor(ix, off, 16);
            if (od < d || (od == d && oi < ix)) { d = od; ix = oi; }
        }
        if (r == 0) {
            int m = (lane >> 4) * 8 + i;
            outIdx[wgRow0 + wave * 16 + m] = ix;
        }
    }
}

__global__ __launch_bounds__(256) void vq_gather_loss(
    const float* __restrict__ x, const float* __restrict__ E,
    const int* __restrict__ idx, float* __restrict__ out,
    unsigned* __restrict__ counts, double* __restrict__ lossAcc) {
    int row = blockIdx.x, t = threadIdx.x;
    int k = idx[row];
    float e  = E[(size_t)k * D_DIM + t];
    float xv = x[(size_t)row * D_DIM + t];
    out[(size_t)row * D_DIM + t] = e;       // straight-through fwd value == quantized
    float d = e - xv;
    float sq = d * d;
#pragma unroll
    for (int off = 16; off > 0; off >>= 1) sq += __shfl_xor(sq, off, 32);
    __shared__ float red[8];
    if ((t & 31) == 0) red[t >> 5] = sq;
    __syncthreads();
    if (t == 0) {
        float s = 0.f;
#pragma unroll
        for (int i = 0; i < 8; ++i) s += red[i];
        atomicAdd(lossAcc, (double)s);      // global_atomic_add_f64
        atomicAdd(&counts[k], 1u);
    }
}

__global__ __launch_bounds__(256) void vq_finalize(const unsigned* __restrict__ counts,
                                                   const double* __restrict__ lossAcc,
                                                   float* __restrict__ outTail) {
    int t = threadIdx.x;
    float ent = 0.f;
    for (int i = t; i < K_EMB; i += 256) {
        float p = (float)counts[i] * (1.0f / (float)N_ROWS);
        ent += p * logf(p + 1e-10f);        // p==0 -> 0, matches reference
    }
#pragma unroll
    for (int off = 16; off > 0; off >>= 1) ent += __shfl_xor(ent, off, 32);
    __shared__ float red[8];
    if ((t & 31) == 0) red[t >> 5] = ent;
    __syncthreads();
    if (t == 0) {
        float s = 0.f;
        for (int i = 0; i < 8; ++i) s += red[i];
        double mse = *lossAcc / (double)((size_t)N_ROWS * D_DIM);
        outTail[0] = (float)(1.25 * mse);   // q_latent + 0.25 * e_latent (equal fwd values)
        outTail[1] = expf(-s);
    }
}

extern "C" void kernel_launch(void* const* d_in, const int* in_sizes, int n_in,
                              void* d_out, int out_size, void* d_ws, size_t ws_size,
                              hipStream_t stream) {
    const float* x = (const float*)d_in[0];     // [16,1024,256]
    const float* E = (const float*)d_in[1];     // [8192,256]
    float* out = (float*)d_out;
    char* ws = (char*)d_ws;
    float*    enorm   = (float*)ws;
    int*      indices = (int*)(ws + 32768);
    unsigned* counts  = (unsigned*)(ws + 98304);
    double*   lossAcc = (double*)(ws + 131072);

    vq_init_ws<<<(K_EMB + 255) / 256, 256, 0, stream>>>(counts, lossAcc);
    vq_enorm<<<K_EMB / 8, 256, 0, stream>>>(E, enorm);

    size_t ldsBytes = (size_t)(ROWS_PER_WG * LDS_STRIDE + 2 * 16 * LDS_STRIDE) * sizeof(float);
    vq_argmin<<<N_ROWS / ROWS_PER_WG, WG_THREADS, ldsBytes, stream>>>(x, E, enorm, indices);

    vq_gather_loss<<<N_ROWS, 256, 0, stream>>>(x, E, indices, out, counts, lossAcc);
    vq_finalize<<<1, 256, 0, stream>>>(counts, lossAcc, out + (size_t)N_ROWS * D_DIM);
}